// Latent_76905684402362
// MI455X (gfx1250) — compile-verified
//
#include <hip/hip_runtime.h>
#include <hip/hip_bf16.h>

typedef __attribute__((ext_vector_type(2))) float v2f;
typedef __attribute__((ext_vector_type(4))) float f4;
typedef __attribute__((ext_vector_type(8))) float v8f;
typedef __attribute__((ext_vector_type(4))) unsigned int u32x4;
typedef __attribute__((ext_vector_type(8))) int i32x8;
typedef __attribute__((ext_vector_type(4))) int i32x4;

#define BDIM 8192
#define HDIM 1024
#define KDIM 512

// ---------------------------------------------------------------------------
// WMMA helper: D = A(16x4) * B(4x16) + C, f32.
// ---------------------------------------------------------------------------
__device__ __forceinline__ v8f wmma_f32(v2f a, v2f b, v8f c) {
    return __builtin_amdgcn_wmma_f32_16x16x4_f32(false, a, false, b, (short)0, c,
                                                 false, false);
}

__device__ __forceinline__ float wave_sum(float v) {
    #pragma unroll
    for (int m = 16; m; m >>= 1) v += __shfl_xor(v, m);
    return v;
}
__device__ __forceinline__ float wave_max(float v) {
    #pragma unroll
    for (int m = 16; m; m >>= 1) v = fmaxf(v, __shfl_xor(v, m));
    return v;
}

// ---------------------------------------------------------------------------
// TDM: DMA one 2-D f32 tile (rows x 32 cols, row stride in elements) from
// global memory into LDS at byte offset lds_off, inserting 4 DWORDs of pad
// after every 32 DWORDs stored -> LDS row stride of 36 floats (144 B), which
// makes the b64 WMMA fragment reads bank-conflict free.
// D# bitfields per CDNA5 ISA.
// ---------------------------------------------------------------------------
__device__ __forceinline__ void tdm_load_tile(unsigned lds_off,
                                              const void* gaddr,
                                              int rows, int rowstride_elems) {
    unsigned long long ga = (unsigned long long)(size_t)gaddr;
    u32x4 g0;
    g0[0] = 1u;                                      // count=1, user mode
    g0[1] = lds_off;                                 // lds_addr (bytes)
    g0[2] = (unsigned)(ga & 0xFFFFFFFFu);            // global_addr[31:0]
    g0[3] = (unsigned)((ga >> 32) & 0x1FFFFFFu)      // global_addr[56:32]
          | (2u << 30);                              // type=2 ("image")
    i32x8 g1;
    g1[0] = (2 << 16)                                // data_size = 4 bytes
          | (1 << 20)                                // pad_enable
          | (4 << 22)                                // pad_interval: 32 DWORDs
          | (3 << 25);                               // pad_amount: 4 DWORDs
    g1[1] = (32 << 16);                              // tensor_dim0 = 32 (lo16)
    g1[2] = (rows << 16);                            // dim0 hi=0 | tensor_dim1 lo16
    g1[3] = (32 << 16);                              // dim1 hi=0 | tile_dim0 = 32
    g1[4] = rows & 0xFFFF;                           // tile_dim1 = rows, tile_dim2=0
    g1[5] = rowstride_elems;                         // tensor_dim0_stride[31:0]
    g1[6] = 0;
    g1[7] = 0;
    i32x4 gz = {0, 0, 0, 0};                         // 2-D tile: groups 2/3 zero
#if __clang_major__ >= 23
    i32x8 gz8 = {0, 0, 0, 0, 0, 0, 0, 0};
    __builtin_amdgcn_tensor_load_to_lds(g0, g1, gz, gz, gz8, 0);
#else
    __builtin_amdgcn_tensor_load_to_lds(g0, g1, gz, gz, 0);
#endif
}

// ---------------------------------------------------------------------------
// Kernel 0: zero the categorical accumulator
// ---------------------------------------------------------------------------
__global__ void zero_f32(float* __restrict__ p, int n) {
    int i = blockIdx.x * blockDim.x + threadIdx.x;
    if (i < n) p[i] = 0.0f;
}

// ---------------------------------------------------------------------------
// Kernel 1: per-cluster precompute. Block = one cluster row k.
// ---------------------------------------------------------------------------
__launch_bounds__(256)
__global__ void prep_clusters(const float* __restrict__ mu_c,
                              const float* __restrict__ ls_c,
                              float* __restrict__ inv_sig,
                              float* __restrict__ mu_inv,
                              float* __restrict__ c2,
                              float* __restrict__ sumls) {
    const int k = blockIdx.x;
    const int t = threadIdx.x;
    const size_t base = (size_t)k * HDIM + 4 * t;
    f4 ls = *(const f4*)&ls_c[base];
    f4 mc = *(const f4*)&mu_c[base];
    f4 inv, mi;
    float c2p = 0.0f, lsp = 0.0f;
    #pragma unroll
    for (int c = 0; c < 4; ++c) {
        float iv = expf(-ls[c]);
        inv[c] = iv;
        float m = mc[c] * iv;
        mi[c] = m;
        c2p += mc[c] * m;
        lsp += ls[c];
    }
    *(f4*)&inv_sig[base] = inv;
    *(f4*)&mu_inv[base]  = mi;

    c2p = wave_sum(c2p);
    lsp = wave_sum(lsp);
    __shared__ float r1[8], r2[8];
    if ((t & 31) == 0) { r1[t >> 5] = c2p; r2[t >> 5] = lsp; }
    __syncthreads();
    if (t == 0) {
        float a = 0.0f, b = 0.0f;
        #pragma unroll
        for (int i = 0; i < 8; ++i) { a += r1[i]; b += r2[i]; }
        c2[k] = a;
        sumls[k] = b;
    }
}

// ---------------------------------------------------------------------------
// Kernel 2: C[b,j] = sum_h X[b,h]*W[j,h] + bias[j]   (NT GEMM, f32 WMMA)
// Block tile 64(M) x 128(N), 8 waves, each wave 16x64 (4 N-tiles).
// Double-buffered TDM staging: while the waves compute chunk i from one LDS
// buffer, the Tensor Data Mover fills the other buffer with chunk i+1.
// K loop unrolled by two chunks so both buffers keep static LDS offsets.
// ---------------------------------------------------------------------------
#define G1_BM 64
#define G1_BN 128
#define G1_KC 32
#define G1_LD 36
#define G1_BUF ((G1_BM + G1_BN) * G1_LD)            // floats per buffer (6912)

__launch_bounds__(256)
__global__ void gemm_xwT_bias(const float* __restrict__ X,
                              const float* __restrict__ W,
                              const float* __restrict__ bias,
                              float* __restrict__ out) {
    extern __shared__ float gsm[];
    float* sA0 = gsm;                               // buf0: A @ 0
    float* sB0 = gsm + G1_BM * G1_LD;               // buf0: B @ 9216 B
    float* sA1 = gsm + G1_BUF;                      // buf1: A @ 27648 B
    float* sB1 = gsm + G1_BUF + G1_BM * G1_LD;      // buf1: B @ 36864 B
    const unsigned bufB = (unsigned)(G1_BUF * sizeof(float));
    const unsigned bOff = (unsigned)(G1_BM * G1_LD * sizeof(float));

    const int m0 = blockIdx.y * G1_BM;
    const int n0 = blockIdx.x * G1_BN;
    const int t = threadIdx.x;
    const int wave = t >> 5;
    const int lane = t & 31;
    const int wm = wave & 3;
    const int wn = wave >> 2;
    const int lrow = lane & 15;
    const int lk = (lane >> 4) << 1;

    v8f acc0 = {}, acc1 = {}, acc2 = {}, acc3 = {};

    auto compute = [&](const float* sAp, const float* sBp) {
        #pragma unroll
        for (int kk = 0; kk < G1_KC; kk += 4) {
            v2f a  = *(const v2f*)&sAp[(wm * 16 + lrow) * G1_LD + kk + lk];
            v2f b0 = *(const v2f*)&sBp[(wn * 64 +  0 + lrow) * G1_LD + kk + lk];
            v2f b1 = *(const v2f*)&sBp[(wn * 64 + 16 + lrow) * G1_LD + kk + lk];
            v2f b2 = *(const v2f*)&sBp[(wn * 64 + 32 + lrow) * G1_LD + kk + lk];
            v2f b3 = *(const v2f*)&sBp[(wn * 64 + 48 + lrow) * G1_LD + kk + lk];
            acc0 = wmma_f32(a, b0, acc0);
            acc1 = wmma_f32(a, b1, acc1);
            acc2 = wmma_f32(a, b2, acc2);
            acc3 = wmma_f32(a, b3, acc3);
        }
    };

    // preload chunk 0 into buf0
    if (wave == 0) {
        tdm_load_tile(0u, &X[(size_t)m0 * HDIM], G1_BM, HDIM);
        tdm_load_tile(bOff, &W[(size_t)n0 * HDIM], G1_BN, HDIM);
    }
    __builtin_amdgcn_s_wait_tensorcnt(0);
    __syncthreads();

    for (int kb = 0; kb < HDIM; kb += 2 * G1_KC) {
        // prefetch chunk kb+KC into buf1 while computing buf0
        if (wave == 0) {
            tdm_load_tile(bufB, &X[(size_t)m0 * HDIM + kb + G1_KC], G1_BM, HDIM);
            tdm_load_tile(bufB + bOff, &W[(size_t)n0 * HDIM + kb + G1_KC], G1_BN, HDIM);
        }
        compute(sA0, sB0);
        __builtin_amdgcn_s_wait_tensorcnt(0);
        __syncthreads();
        // prefetch chunk kb+2*KC into buf0 while computing buf1
        if (wave == 0 && kb + 2 * G1_KC < HDIM) {
            tdm_load_tile(0u, &X[(size_t)m0 * HDIM + kb + 2 * G1_KC], G1_BM, HDIM);
            tdm_load_tile(bOff, &W[(size_t)n0 * HDIM + kb + 2 * G1_KC], G1_BN, HDIM);
        }
        compute(sA1, sB1);
        __builtin_amdgcn_s_wait_tensorcnt(0);
        __syncthreads();
    }

    const int rowBase = m0 + wm * 16 + ((lane >> 4) << 3);
    const int colBase = n0 + wn * 64 + lrow;
    v8f accs[4] = {acc0, acc1, acc2, acc3};
    #pragma unroll
    for (int nt = 0; nt < 4; ++nt) {
        const int col = colBase + nt * 16;
        const float bv = bias[col];
        #pragma unroll
        for (int v = 0; v < 8; ++v)
            out[(size_t)(rowBase + v) * HDIM + col] = accs[nt][v] + bv;
    }
}

// ---------------------------------------------------------------------------
// Kernel 3: z = mu_z + exp(0.5*ls_z)*eps  -> d_out[0 .. B*H)
// ---------------------------------------------------------------------------
__launch_bounds__(256)
__global__ void make_z(const float* __restrict__ mu,
                       const float* __restrict__ ls,
                       const float* __restrict__ eps,
                       float* __restrict__ z, int n4) {
    int i = blockIdx.x * blockDim.x + threadIdx.x;
    if (i >= n4) return;
    f4 m = ((const f4*)mu)[i];
    f4 l = ((const f4*)ls)[i];
    f4 e = ((const f4*)eps)[i];
    f4 r;
    #pragma unroll
    for (int c = 0; c < 4; ++c) r[c] = m[c] + expf(0.5f * l[c]) * e[c];
    ((f4*)z)[i] = r;
}

// ---------------------------------------------------------------------------
// Kernel 4: fused mahalanobis GEMMs + softmax + latent loss + cate partials.
// One block = 16 batch rows, 512 threads (16 waves). Wave w owns clusters
// [w*32, w*32+32). A matrices: z^2, mu^2+exp(ls), z, mu. B fragments come
// straight from L2-resident inv_sig / mu_inv. A-tile global loads are
// software-pipelined one chunk ahead so they retire under the WMMA loop.
// Dynamic LDS = 64KB, reused between A staging and output matrices.
// ---------------------------------------------------------------------------
#define G2_KC 64
#define G2_LD 68

__launch_bounds__(512)
__global__ void fused_latent(const float* __restrict__ mu_z,
                             const float* __restrict__ ls_z,
                             const float* __restrict__ zmat,
                             const float* __restrict__ inv_sig,
                             const float* __restrict__ mu_inv,
                             const float* __restrict__ c2,
                             const float* __restrict__ sumls,
                             float* __restrict__ cate_acc,
                             float* __restrict__ latent_out) {
    extern __shared__ float lds[];
    float* sA = lds;
    const int m0 = blockIdx.x * 16;
    const int t = threadIdx.x;
    const int wave = t >> 5;
    const int lane = t & 31;
    const int lrow = lane & 15;
    const int lk = (lane >> 4) << 1;

    v8f aZsq0 = {}, aZsq1 = {}, aAp0 = {}, aAp1 = {};
    v8f aZ0 = {}, aZ1 = {}, aMu0 = {}, aMu1 = {};

    const int srow = t >> 5;
    const int scol = lane << 1;
    const int c0 = wave * 32 + lrow;
    const int c1 = c0 + 16;

    const size_t gbase = (size_t)(m0 + srow) * HDIM + scol;
    // prologue: load chunk 0 staging registers
    v2f mv = *(const v2f*)&mu_z[gbase];
    v2f lv = *(const v2f*)&ls_z[gbase];
    v2f zv = *(const v2f*)&zmat[gbase];

    for (int kb = 0; kb < HDIM; kb += G2_KC) {
        // ---- store staged A tiles from registers ----
        {
            const int lb = srow * G2_LD + scol;
            v2f zsq = {zv[0] * zv[0], zv[1] * zv[1]};
            v2f ap  = {mv[0] * mv[0] + expf(lv[0]), mv[1] * mv[1] + expf(lv[1])};
            *(v2f*)&sA[0 * 16 * G2_LD + lb] = zsq;
            *(v2f*)&sA[1 * 16 * G2_LD + lb] = ap;
            *(v2f*)&sA[2 * 16 * G2_LD + lb] = zv;
            *(v2f*)&sA[3 * 16 * G2_LD + lb] = mv;
        }
        __syncthreads();
        // ---- issue next chunk's global loads (overlap with WMMA loop) ----
        if (kb + G2_KC < HDIM) {
            const size_t g = gbase + kb + G2_KC;
            mv = *(const v2f*)&mu_z[g];
            lv = *(const v2f*)&ls_z[g];
            zv = *(const v2f*)&zmat[g];
        }
        #pragma unroll 4
        for (int kk = 0; kk < G2_KC; kk += 4) {
            const int ao = lrow * G2_LD + kk + lk;
            v2f fZsq = *(const v2f*)&sA[0 * 16 * G2_LD + ao];
            v2f fAp  = *(const v2f*)&sA[1 * 16 * G2_LD + ao];
            v2f fZ   = *(const v2f*)&sA[2 * 16 * G2_LD + ao];
            v2f fMu  = *(const v2f*)&sA[3 * 16 * G2_LD + ao];
            const size_t bg0 = (size_t)c0 * HDIM + kb + kk + lk;
            const size_t bg1 = (size_t)c1 * HDIM + kb + kk + lk;
            v2f bI0 = *(const v2f*)&inv_sig[bg0];
            v2f bI1 = *(const v2f*)&inv_sig[bg1];
            v2f bM0 = *(const v2f*)&mu_inv[bg0];
            v2f bM1 = *(const v2f*)&mu_inv[bg1];
            aZsq0 = wmma_f32(fZsq, bI0, aZsq0);
            aZsq1 = wmma_f32(fZsq, bI1, aZsq1);
            aAp0  = wmma_f32(fAp,  bI0, aAp0);
            aAp1  = wmma_f32(fAp,  bI1, aAp1);
            aZ0   = wmma_f32(fZ,   bM0, aZ0);
            aZ1   = wmma_f32(fZ,   bM1, aZ1);
            aMu0  = wmma_f32(fMu,  bM0, aMu0);
            aMu1  = wmma_f32(fMu,  bM1, aMu1);
        }
        __syncthreads();
    }

    float* sOut = lds;
    const int rowB = (lane >> 4) << 3;
    const int colB = wave * 32 + lrow;
    #pragma unroll
    for (int v = 0; v < 8; ++v) {
        const int r = rowB + v;
        sOut[0 * 16 * 512 + r * 512 + colB]      = aZsq0[v] - 2.0f * aZ0[v];
        sOut[0 * 16 * 512 + r * 512 + colB + 16] = aZsq1[v] - 2.0f * aZ1[v];
        sOut[1 * 16 * 512 + r * 512 + colB]      = aAp0[v] - 2.0f * aMu0[v];
        sOut[1 * 16 * 512 + r * 512 + colB + 16] = aAp1[v] - 2.0f * aMu1[v];
    }
    __syncthreads();

    {
        const int r = wave;
        float* mzrow = &sOut[0 * 16 * 512 + r * 512];
        const float* avrow = &sOut[1 * 16 * 512 + r * 512];

        float lmax = -3.4e38f;
        for (int k = lane; k < KDIM; k += 32)
            lmax = fmaxf(lmax, -(mzrow[k] + c2[k]));
        lmax = wave_max(lmax);

        float sexp = 0.0f;
        for (int k = lane; k < KDIM; k += 32)
            sexp += expf(-(mzrow[k] + c2[k]) - lmax);
        sexp = wave_sum(sexp);
        const float rinv = 1.0f / sexp;

        float sum_ = 0.0f;
        for (int k = lane; k < KDIM; k += 32) {
            const float att = expf(-(mzrow[k] + c2[k]) - lmax) * rinv;
            const float avg = (sumls[k] + avrow[k] + c2[k]) * (1.0f / HDIM);
            sum_ += att * avg;
            mzrow[k] = att;
        }
        sum_ = wave_sum(sum_);

        float ms = 0.0f;
        const float* lsr = &ls_z[(size_t)(m0 + r) * HDIM];
        for (int h = lane; h < HDIM; h += 32) ms += 1.0f + lsr[h];
        ms = wave_sum(ms);

        if (lane == 0)
            latent_out[m0 + r] = 0.5f * sum_ - 0.5f * (ms * (1.0f / HDIM));
    }
    __syncthreads();

    {
        float s = 0.0f;
        #pragma unroll
        for (int rr = 0; rr < 16; ++rr) s += sOut[0 * 16 * 512 + rr * 512 + t];
        atomicAdd(&cate_acc[t], s);
    }
}

// ---------------------------------------------------------------------------
// Kernel 5: batch_cate_loss = mean_k( (acc/B) * log(acc/B) )
// ---------------------------------------------------------------------------
__launch_bounds__(512)
__global__ void cate_finalize(const float* __restrict__ cate_acc,
                              float* __restrict__ out_scalar) {
    const int t = threadIdx.x;
    float v = cate_acc[t] * (1.0f / BDIM);
    float term = v * logf(v);
    term = wave_sum(term);
    __shared__ float red[16];
    if ((t & 31) == 0) red[t >> 5] = term;
    __syncthreads();
    if (t == 0) {
        float s = 0.0f;
        #pragma unroll
        for (int i = 0; i < 16; ++i) s += red[i];
        *out_scalar = s * (1.0f / KDIM);
    }
}

// ---------------------------------------------------------------------------
extern "C" void kernel_launch(void* const* d_in, const int* in_sizes, int n_in,
                              void* d_out, int out_size, void* d_ws, size_t ws_size,
                              hipStream_t stream) {
    const float* X    = (const float*)d_in[0];
    const float* W_mu = (const float*)d_in[1];
    const float* b_mu = (const float*)d_in[2];
    const float* W_ls = (const float*)d_in[3];
    const float* b_ls = (const float*)d_in[4];
    const float* mu_c = (const float*)d_in[5];
    const float* ls_c = (const float*)d_in[6];
    const float* eps  = (const float*)d_in[7];

    float* out    = (float*)d_out;
    float* z      = out;
    float* latent = out + (size_t)BDIM * HDIM;
    float* cate   = latent + BDIM;

    float* ws       = (float*)d_ws;
    float* mu_z     = ws;
    float* ls_z     = mu_z + (size_t)BDIM * HDIM;
    float* inv_sig  = ls_z + (size_t)BDIM * HDIM;
    float* mu_inv   = inv_sig + (size_t)KDIM * HDIM;
    float* c2       = mu_inv + (size_t)KDIM * HDIM;
    float* sumls    = c2 + KDIM;
    float* cate_acc = sumls + KDIM;

    zero_f32<<<1, 512, 0, stream>>>(cate_acc, KDIM);
    prep_clusters<<<KDIM, 256, 0, stream>>>(mu_c, ls_c, inv_sig, mu_inv, c2, sumls);

    dim3 g1(HDIM / G1_BN, BDIM / G1_BM);
    const int g1_lds = 2 * G1_BUF * sizeof(float);              // 55296 B
    gemm_xwT_bias<<<g1, 256, g1_lds, stream>>>(X, W_mu, b_mu, mu_z);
    gemm_xwT_bias<<<g1, 256, g1_lds, stream>>>(X, W_ls, b_ls, ls_z);

    const int n4 = BDIM * HDIM / 4;
    make_z<<<n4 / 256, 256, 0, stream>>>(mu_z, ls_z, eps, z, n4);

    fused_latent<<<BDIM / 16, 512, 2 * 16 * 512 * sizeof(float), stream>>>(
        mu_z, ls_z, z, inv_sig, mu_inv, c2, sumls, cate_acc, latent);

    cate_finalize<<<1, 512, 0, stream>>>(cate_acc, cate);
}